// DBHMM_48223892799793
// MI455X (gfx1250) — compile-verified
//
#include <hip/hip_runtime.h>
#include <math.h>

#define VOCAB  32000
#define NS     512
#define BATCH  64
#define MAXLEN 512
#define RWG    4              // cooperating workgroups in the recurrence
#define RROWS  (NS / RWG)     // 128 rows of M per WG (256 KB LDS slice)

typedef float v2f __attribute__((ext_vector_type(2)));
typedef float v8f __attribute__((ext_vector_type(8)));

// ---------- helpers ----------
__device__ __forceinline__ void lse_merge(float& mx, float& sm, float omx, float osm) {
    float nmx = fmaxf(mx, omx);
    if (nmx == -INFINITY) { mx = nmx; sm = sm + osm; return; }
    sm = sm * expf(mx - nmx) + osm * expf(omx - nmx);
    mx = nmx;
}

__device__ __forceinline__ void lse_accum(float& mx, float& sm, float v) {
    if (v > mx) { sm = sm * expf(mx - v) + 1.0f; mx = v; }
    else        { sm += expf(v - mx); }
}

// deterministic 512-thread block logsumexp (fixed tree order)
__device__ __forceinline__ float block_lse512(float x, float* red) {
    const int j = threadIdx.x;
    red[j] = x;
    __syncthreads();
    for (int s = 256; s > 0; s >>= 1) {
        if (j < s) red[j] = fmaxf(red[j], red[j + s]);
        __syncthreads();
    }
    float mx = red[0];
    __syncthreads();
    red[j] = expf(x - mx);
    __syncthreads();
    for (int s = 256; s > 0; s >>= 1) {
        if (j < s) red[j] += red[j + s];
        __syncthreads();
    }
    float lse = mx + logf(red[0]);
    __syncthreads();
    return lse;
}

// ---------- kernel 1a: per-column logsumexp over vocab (chunked) ----------
__global__ void col_lse_partial(const float* __restrict__ emb,
                                float* __restrict__ pmax, float* __restrict__ psum) {
    const int c     = blockIdx.x * 32 + threadIdx.x;     // column 0..511
    const int chunk = blockIdx.y;                        // 0..31
    const int r0    = chunk * (VOCAB / 32);              // 1000-row chunk
    float mx = -INFINITY, sm = 0.0f;
    for (int r = r0 + threadIdx.y; r < r0 + (VOCAB / 32); r += 8) {
        lse_accum(mx, sm, emb[(size_t)r * NS + c]);      // 128B coalesced across x
    }
    __shared__ float smx[8][32];
    __shared__ float ssm[8][32];
    smx[threadIdx.y][threadIdx.x] = mx;
    ssm[threadIdx.y][threadIdx.x] = sm;
    __syncthreads();
    if (threadIdx.y == 0) {
        for (int y = 1; y < 8; ++y) lse_merge(mx, sm, smx[y][threadIdx.x], ssm[y][threadIdx.x]);
        pmax[chunk * NS + c] = mx;
        psum[chunk * NS + c] = sm;
    }
}

// ---------- kernel 1b: combine 32 chunks per column ----------
__global__ void col_lse_combine(const float* __restrict__ pmax, const float* __restrict__ psum,
                                float* __restrict__ lse_col) {
    const int c = threadIdx.x;
    float mx = -INFINITY, sm = 0.0f;
    for (int k = 0; k < 32; ++k) lse_merge(mx, sm, pmax[k * NS + c], psum[k * NS + c]);
    lse_col[c] = mx + logf(sm);
}

// ---------- kernel 2: M = d1 @ d2^T via V_WMMA_F32_16X16X4_F32 ----------
__global__ void wmma_gemm_d1d2t(const float* __restrict__ d1, const float* __restrict__ d2,
                                float* __restrict__ M) {
    const int ti   = blockIdx.y * 16;
    const int tj   = blockIdx.x * 16;
    const int lane = threadIdx.x;              // 0..31 (wave32)
    const int half = lane >> 4;                // 0: K=0,1 ; 1: K=2,3
    const int l15  = lane & 15;

    const float* arow = d1 + (size_t)(ti + l15) * NS;
    const float* brow = d2 + (size_t)(tj + l15) * NS;

    v8f c = {};
#pragma unroll 8
    for (int k = 0; k < NS; k += 4) {
        const int ka = k + 2 * half;
        v2f a = *(const v2f*)(arow + ka);
        v2f b = *(const v2f*)(brow + ka);
        c = __builtin_amdgcn_wmma_f32_16x16x4_f32(false, a, false, b, (short)0, c, false, false);
    }
#pragma unroll
    for (int r = 0; r < 8; ++r) {
        M[(size_t)(ti + r + 8 * half) * NS + (tj + l15)] = c[r];
    }
}

// ---------- kernel 3: cooperative recurrence, M LDS-resident across 4 WGPs ----------
// grid = 4 blocks x 512 threads; each block stages a 128x512 slice of M (256KB) in LDS.
// Per step: local 128-row partial matvec from LDS, 2KB partial exchange via L2,
// monotonic-counter device barrier (double-buffered partials), redundant local logsumexp.
__global__ void __launch_bounds__(512, 1)
recurrence_kernel(const float* __restrict__ begin, const float* __restrict__ M,
                  float* __restrict__ hidden, float* __restrict__ xpart,
                  unsigned int* __restrict__ bar) {
    __shared__ float Mlds[RROWS * NS];   // 256 KB slice
    __shared__ float cur[NS];
    __shared__ float red[NS];
    const int j  = threadIdx.x;
    const int wg = blockIdx.x;           // 0..3
    const int ibase = wg * RROWS;

    // stage this WG's slice of M into LDS (b128 copies)
    const float4* Msrc = (const float4*)(M + (size_t)ibase * NS);
    float4* Mdst = (float4*)Mlds;
    for (int idx = j; idx < RROWS * NS / 4; idx += 512) Mdst[idx] = Msrc[idx];

    // m0 = log_softmax(begin), computed redundantly per WG (bitwise identical)
    float x   = begin[j];
    float lse = block_lse512(x, red);    // barriers here also cover LDS staging
    float m   = x - lse;
    cur[j]    = m;
    if (wg == 0) hidden[j] = m;
    __syncthreads();

    for (int k = 1; k <= 256; ++k) {
        const int buf = (k & 1) * (RWG * NS);     // double buffer: one barrier/step
        float acc = 0.0f;
#pragma unroll 8
        for (int i = 0; i < RROWS; ++i) {
            acc = fmaf(cur[ibase + i], Mlds[i * NS + j], acc);  // conflict-free LDS
        }
        xpart[buf + wg * NS + j] = acc;
        __threadfence();                          // partials visible at device scope
        __syncthreads();
        if (j == 0) {
            __hip_atomic_fetch_add(bar, 1u, __ATOMIC_ACQ_REL, __HIP_MEMORY_SCOPE_AGENT);
            while (__hip_atomic_load(bar, __ATOMIC_ACQUIRE, __HIP_MEMORY_SCOPE_AGENT)
                   < (unsigned)(RWG * k)) {
                __builtin_amdgcn_s_sleep(1);
            }
        }
        __syncthreads();
        float tot = 0.0f;
#pragma unroll
        for (int w = 0; w < RWG; ++w) {           // fixed order -> deterministic
            tot += __hip_atomic_load(&xpart[buf + w * NS + j],
                                     __ATOMIC_RELAXED, __HIP_MEMORY_SCOPE_AGENT);
        }
        float l  = block_lse512(tot, red);
        float nv = tot - l;                       // f^k(m0)
        if (wg == 0) {
            hidden[(size_t)(2 * k - 1) * NS + j] = nv;
            if (2 * k < MAXLEN) hidden[(size_t)(2 * k) * NS + j] = nv;
        }
        __syncthreads();
        cur[j] = nv;
        __syncthreads();
    }
}

// ---------- kernel 4: per-(t,b) logsumexp, one wave32 per pair ----------
__global__ void ppl_partial_kernel(const int* __restrict__ sent, const float* __restrict__ masks,
                                   const float* __restrict__ emb, const float* __restrict__ hidden,
                                   const float* __restrict__ lse_col, float* __restrict__ partial) {
    const int lane = threadIdx.x & 31;
    const int wave = threadIdx.x >> 5;
    const int p    = blockIdx.x * 8 + wave;    // 0..32767
    const int t    = p >> 6;
    const int b    = p & 63;
    const int row  = sent[b * MAXLEN + t];     // sentences[b, t]
    const float* erow = emb + (size_t)row * NS;
    const float* hrow = hidden + (size_t)t * NS;

    __builtin_prefetch(erow + lane, 0, 0);     // global_prefetch_b8 on the gather row

    float mx = -INFINITY, sm = 0.0f;
#pragma unroll
    for (int jj = 0; jj < NS; jj += 32) {
        const int j = jj + lane;
        lse_accum(mx, sm, hrow[j] + erow[j] - lse_col[j]);
    }
#pragma unroll
    for (int off = 16; off > 0; off >>= 1) {   // wave32 butterfly
        float omx = __shfl_xor(mx, off, 32);
        float osm = __shfl_xor(sm, off, 32);
        lse_merge(mx, sm, omx, osm);
    }
    if (lane == 0) partial[p] = (mx + logf(sm)) * masks[b * MAXLEN + t];
}

// ---------- kernel 5: deterministic final sum ----------
__global__ void final_sum_kernel(const float* __restrict__ partial, float* __restrict__ out) {
    __shared__ float red[512];
    const int j = threadIdx.x;
    float s = 0.0f;
    for (int i = j; i < MAXLEN * BATCH; i += 512) s += partial[i];
    red[j] = s;
    __syncthreads();
    for (int st = 256; st > 0; st >>= 1) {
        if (j < st) red[j] += red[j + st];
        __syncthreads();
    }
    if (j == 0) out[0] = red[0];
}

extern "C" void kernel_launch(void* const* d_in, const int* in_sizes, int n_in,
                              void* d_out, int out_size, void* d_ws, size_t ws_size,
                              hipStream_t stream) {
    (void)in_sizes; (void)n_in; (void)out_size; (void)ws_size;
    const int*   sent  = (const int*)d_in[0];    // [64, 512] int32
    const float* masks = (const float*)d_in[1];  // [64, 512]
    const float* emb   = (const float*)d_in[2];  // [32000, 512]
    const float* d1    = (const float*)d_in[3];  // [512, 512]
    const float* d2    = (const float*)d_in[4];  // [512, 512]
    const float* begin = (const float*)d_in[5];  // [512]
    float* out = (float*)d_out;

    float* ws      = (float*)d_ws;
    float* lse_col = ws;                         // 512 (padded to 1024)
    float* Mmat    = ws + 1024;                  // 512*512
    float* hidden  = Mmat + NS * NS;             // 512*512
    float* partial = hidden + NS * NS;           // 32768
    float* pmax    = partial + MAXLEN * BATCH;   // 32*512
    float* psum    = pmax + 32 * NS;             // 32*512
    float* xpart   = psum + 32 * NS;             // 2*RWG*512 partial buffers
    unsigned int* bar = (unsigned int*)(xpart + 2 * RWG * NS);

    hipMemsetAsync(bar, 0, sizeof(unsigned int), stream);   // capture-safe barrier init

    col_lse_partial   <<<dim3(16, 32), dim3(32, 8), 0, stream>>>(emb, pmax, psum);
    col_lse_combine   <<<1, 512, 0, stream>>>(pmax, psum, lse_col);
    wmma_gemm_d1d2t   <<<dim3(32, 32), 32, 0, stream>>>(d1, d2, Mmat);
    recurrence_kernel <<<RWG, 512, 0, stream>>>(begin, Mmat, hidden, xpart, bar);
    ppl_partial_kernel<<<(MAXLEN * BATCH) / 8, 256, 0, stream>>>(sent, masks, emb, hidden, lse_col, partial);
    final_sum_kernel  <<<1, 512, 0, stream>>>(partial, out);
}